// Modality_2_Attention_45767171506398
// MI455X (gfx1250) — compile-verified
//
#include <hip/hip_runtime.h>
#include <hip/hip_bf16.h>
#include <math.h>

typedef float v2f __attribute__((ext_vector_type(2)));
typedef float v8f __attribute__((ext_vector_type(8)));

#define BB 16
#define NN 32768
#define CC 32
#define NPC (NN / 32)   // 1024 rows per pool chunk

// workspace layout (float offsets)
#define WS_M1   0                    // [16][32][32]
#define WS_M2   16384                // [16][32][32]
#define WS_QT1  32768                // [b*2+h][o][p]  (transposed Q1)
#define WS_QT2  65536                // [b*2+h][o][p]
#define WS_PART 98304                // part1[32], part2[32]
#define WS_MAT  98432                // [3][16][32][32]  fused matrices M1,M2,M3

// ---------------------------------------------------------------------------
// Kernel 1: adaptive max pool  [B,N,32] -> [B,32,32]  (both emb tensors)
// ---------------------------------------------------------------------------
__global__ __launch_bounds__(256) void pool_kernel(const float* __restrict__ emb1,
                                                   const float* __restrict__ emb2,
                                                   float* __restrict__ ws) {
  __shared__ float red[256];
  int e = blockIdx.x;
  int t = e >> 9;            // 0: emb1, 1: emb2   (512 blocks each)
  int rem = e & 511;
  int b = rem >> 5;
  int p = rem & 31;
  const float* src = (t == 0) ? emb1 : emb2;
  float* dst = ws + ((t == 0) ? WS_M1 : WS_M2);
  int c = threadIdx.x & 31;
  int r = threadIdx.x >> 5;  // 0..7
  const float* base = src + ((size_t)b * NN + (size_t)p * NPC) * CC;
  float acc = -INFINITY;
  for (int q = r; q < NPC; q += 8) acc = fmaxf(acc, base[(size_t)q * CC + c]);
  red[threadIdx.x] = acc;
  __syncthreads();
  if (r == 0) {
    float m = red[c];
#pragma unroll
    for (int k = 1; k < 8; ++k) m = fmaxf(m, red[k * 32 + c]);
    dst[(b * 32 + p) * 32 + c] = m;
  }
}

// ---------------------------------------------------------------------------
// Kernel 2: per-(b,h) query projections Q = m @ Wq^T, stored transposed
// as Qt[bh][o][p]; also deterministic per-wave sum-of-squares partials.
// grid = 32 blocks (b,h), block = 32 threads (one wave, lane = o)
// ---------------------------------------------------------------------------
__global__ __launch_bounds__(32) void proj_kernel(const float* __restrict__ Wq1,
                                                  const float* __restrict__ Wq2,
                                                  float* __restrict__ ws) {
  int bh = blockIdx.x;
  int b = bh >> 1, h = bh & 1;
  int o = threadIdx.x;  // 0..31
  __shared__ float sm1[32][32], sm2[32][32];
  const float* m1 = ws + WS_M1 + b * 1024;
  const float* m2 = ws + WS_M2 + b * 1024;
#pragma unroll 4
  for (int p = 0; p < 32; ++p) {  // coalesced: lane o loads element o of row p
    sm1[p][o] = m1[p * 32 + o];
    sm2[p][o] = m2[p * 32 + o];
  }
  __syncthreads();
  float w1[32], w2[32];
#pragma unroll
  for (int cc = 0; cc < 32; ++cc) {
    w1[cc] = Wq1[(h * 32 + o) * 32 + cc];
    w2[cc] = Wq2[(h * 32 + o) * 32 + cc];
  }
  float ss1 = 0.f, ss2 = 0.f;
  float* qt1 = ws + WS_QT1 + (bh * 32 + o) * 32;
  float* qt2 = ws + WS_QT2 + (bh * 32 + o) * 32;
  for (int p = 0; p < 32; ++p) {
    float a1 = 0.f, a2 = 0.f;
#pragma unroll
    for (int cc = 0; cc < 32; ++cc) {
      a1 = fmaf(sm1[p][cc], w1[cc], a1);
      a2 = fmaf(sm2[p][cc], w2[cc], a2);
    }
    qt1[p] = a1;
    qt2[p] = a2;
    ss1 = fmaf(a1, a1, ss1);
    ss2 = fmaf(a2, a2, ss2);
  }
#pragma unroll
  for (int off = 16; off > 0; off >>= 1) {
    ss1 += __shfl_xor(ss1, off, 32);
    ss2 += __shfl_xor(ss2, off, 32);
  }
  if (o == 0) {
    ws[WS_PART + bh] = ss1;
    ws[WS_PART + 32 + bh] = ss2;
  }
}

// ---------------------------------------------------------------------------
// Kernel 3: scores -> instance norm -> softmax -> head mean -> M_k = p_k@Wout^T
// Also writes out_s* outputs. grid = 16 (b), block = 64 (wave = head)
// ---------------------------------------------------------------------------
__global__ __launch_bounds__(64) void attn_kernel(const float* __restrict__ Wout1,
                                                  const float* __restrict__ Wout2,
                                                  float* __restrict__ ws,
                                                  float* __restrict__ outs) {
  int b = blockIdx.x;
  int tid = threadIdx.x;
  int h = tid >> 5;
  int lane = tid & 31;
  __shared__ float Q1[2][32][32];  // [h][i][p]  (row i of Q^T = column i of Q)
  __shared__ float Q2[2][32][32];
  __shared__ float accp[3][32][32];  // softmax head-mean
  __shared__ float acca[3][32][32];  // |s| head-sum
  __shared__ float tr[2][32][33];    // transpose scratch per wave

  for (int i = tid; i < 3 * 1024; i += 64) {
    (&accp[0][0][0])[i] = 0.f;
    (&acca[0][0][0])[i] = 0.f;
  }
  const float* g1 = ws + WS_QT1 + ((b * 2 + h) * 32 + lane) * 32;
  const float* g2 = ws + WS_QT2 + ((b * 2 + h) * 32 + lane) * 32;
  float c1[32], c2[32];  // this lane's column (j = lane) of Q1 / Q2
#pragma unroll
  for (int p = 0; p < 32; ++p) {
    c1[p] = g1[p];
    c2[p] = g2[p];
    Q1[h][lane][p] = c1[p];
    Q2[h][lane][p] = c2[p];
  }
  __syncthreads();

  float t1 = 0.f, t2 = 0.f;
  for (int k = 0; k < 32; ++k) {  // fixed-order (deterministic) norm reduce
    t1 += ws[WS_PART + k];
    t2 += ws[WS_PART + 32 + k];
  }
  float n1 = sqrtf(t1), n2 = sqrtf(t2);
  const float isk = 0.17677669529663687f;  // 1/sqrt(32)
  float scl01 = isk / (n1 + n2);
  float scl2 = isk / (2.f * n1);

#pragma unroll
  for (int m = 0; m < 3; ++m) {
    float scl = (m == 2) ? scl2 : scl01;
    float s[32];
#pragma unroll
    for (int i = 0; i < 32; ++i) {
      float a = 0.f;
      if (m == 0) {
#pragma unroll
        for (int p = 0; p < 32; ++p) a = fmaf(Q1[h][i][p], c2[p], a);  // sAB
      } else if (m == 1) {
#pragma unroll
        for (int p = 0; p < 32; ++p) a = fmaf(Q2[h][i][p], c1[p], a);  // sBA
      } else {
#pragma unroll
        for (int p = 0; p < 32; ++p) a = fmaf(Q1[h][i][p], c1[p], a);  // sAA
      }
      s[i] = a * scl;
      atomicAdd(&acca[m][i][lane], fabsf(s[i]));  // 2 adds, commutative -> stable
    }
    // instance norm over the whole 32x32 matrix
    float ls = 0.f, lq = 0.f;
#pragma unroll
    for (int i = 0; i < 32; ++i) {
      ls += s[i];
      lq = fmaf(s[i], s[i], lq);
    }
#pragma unroll
    for (int off = 16; off > 0; off >>= 1) {
      ls += __shfl_xor(ls, off, 32);
      lq += __shfl_xor(lq, off, 32);
    }
    float mean = ls * (1.f / 1024.f);
    float var = lq * (1.f / 1024.f) - mean * mean;
    float rinv = rsqrtf(var + 1e-5f);
#pragma unroll
    for (int i = 0; i < 32; ++i) tr[h][i][lane] = (s[i] - mean) * rinv;
    __syncthreads();
    // softmax along j (this lane now owns row i = lane)
    float r[32];
    float mx = -INFINITY;
#pragma unroll
    for (int j = 0; j < 32; ++j) {
      r[j] = tr[h][lane][j];
      mx = fmaxf(mx, r[j]);
    }
    float sum = 0.f;
#pragma unroll
    for (int j = 0; j < 32; ++j) {
      r[j] = expf(r[j] - mx);
      sum += r[j];
    }
    float inv = 0.5f / sum;  // includes head-mean factor
#pragma unroll
    for (int j = 0; j < 32; ++j) atomicAdd(&accp[m][lane][j], r[j] * inv);
    __syncthreads();
  }

  // out_s outputs:  mean_h(|s|) * 10  ==  acca * 5
  for (int i = tid; i < 3 * 1024; i += 64) {
    int m = i >> 10;
    int ij = i & 1023;
    outs[m * (BB * 1024) + b * 1024 + ij] = (&acca[0][0][0])[i] * 5.0f;
  }
  // fused matrices  M_m[j][o] = sum_c p_m[j][c] * Wout[o][c]
  float* mats = ws + WS_MAT;
  for (int w = tid; w < 96; w += 64) {
    int m = w >> 5;
    int o = w & 31;
    const float* W = (m == 0) ? Wout1 : Wout2;  // O3 uses Wout2 (faithful)
    float wr[32];
#pragma unroll
    for (int cc = 0; cc < 32; ++cc) wr[cc] = W[o * 32 + cc];
    for (int j = 0; j < 32; ++j) {
      float a = 0.f;
#pragma unroll
      for (int cc = 0; cc < 32; ++cc) a = fmaf(accp[m][j][cc], wr[cc], a);
      mats[m * (BB * 1024) + b * 1024 + j * 32 + o] = a;
    }
  }
}

// ---------------------------------------------------------------------------
// Kernel 4: streaming WMMA GEMM.  O1 = emb1@M1, O2 = emb2@M2, O3 = emb1@M3.
// V_WMMA_F32_16X16X4_F32; 16-row tiles, K=32 (8 wmma per 16x16 output tile).
// grid = 16 b * 64 blocks, 8 waves/block, 4 tiles/wave -> 2048 tiles per b.
// ---------------------------------------------------------------------------
__global__ __launch_bounds__(256) void gemm_kernel(const float* __restrict__ emb1,
                                                   const float* __restrict__ emb2,
                                                   const float* __restrict__ ws,
                                                   float* __restrict__ out) {
  int b = blockIdx.x >> 6;
  int blkInB = blockIdx.x & 63;
  int wave = threadIdx.x >> 5;
  int lane = threadIdx.x & 31;
  __shared__ float sM[3][32][32];
  const float* mats = ws + WS_MAT;
  for (int i = threadIdx.x; i < 3072; i += 256) {
    int m = i >> 10;
    (&sM[0][0][0])[i] = mats[m * (BB * 1024) + b * 1024 + (i & 1023)];
  }
  __syncthreads();

  int kb = (lane >> 4) * 2;  // K sub-offset for this lane half
  int nl = lane & 15;        // N within col tile
  // B fragments: Bf[v][lane] = M[kk*4 + kb + v][nt*16 + nl]
  v2f Bf[3][2][8];
#pragma unroll
  for (int m = 0; m < 3; ++m)
#pragma unroll
    for (int nt = 0; nt < 2; ++nt)
#pragma unroll
      for (int kk = 0; kk < 8; ++kk) {
        v2f f;
        f.x = sM[m][kk * 4 + kb + 0][nt * 16 + nl];
        f.y = sM[m][kk * 4 + kb + 1][nt * 16 + nl];
        Bf[m][nt][kk] = f;
      }

  const v8f vzero = {0.f, 0.f, 0.f, 0.f, 0.f, 0.f, 0.f, 0.f};
  int tile0 = (blkInB * 8 + wave) * 4;
  for (int t = 0; t < 4; ++t) {
    int row0 = (tile0 + t) * 16;
    const float* a1 = emb1 + ((size_t)b * NN + row0 + nl) * CC + kb;
    const float* a2 = emb2 + ((size_t)b * NN + row0 + nl) * CC + kb;
    v8f acc[3][2];
#pragma unroll
    for (int m = 0; m < 3; ++m)
#pragma unroll
      for (int nt = 0; nt < 2; ++nt) acc[m][nt] = vzero;

#pragma unroll
    for (int kk = 0; kk < 8; ++kk) {
      v2f A1 = *(const v2f*)(a1 + kk * 4);  // A[v] = emb[row][kk*4+kb+v]
      v2f A2 = *(const v2f*)(a2 + kk * 4);
#pragma unroll
      for (int nt = 0; nt < 2; ++nt) {
        acc[0][nt] = __builtin_amdgcn_wmma_f32_16x16x4_f32(
            false, A1, false, Bf[0][nt][kk], (short)0, acc[0][nt], false, false);
        acc[1][nt] = __builtin_amdgcn_wmma_f32_16x16x4_f32(
            false, A2, false, Bf[1][nt][kk], (short)0, acc[1][nt], false, false);
        acc[2][nt] = __builtin_amdgcn_wmma_f32_16x16x4_f32(
            false, A1, false, Bf[2][nt][kk], (short)0, acc[2][nt], false, false);
      }
    }
    // store: C[v][lane] = D[(lane>>4)*8 + v][nt*16 + nl]
    size_t obase = ((size_t)b * NN + row0) * CC;
    int rhalf = (lane >> 4) * 8;
#pragma unroll
    for (int m = 0; m < 3; ++m) {
      float* ob = out + (size_t)m * ((size_t)BB * NN * CC) + obase;
#pragma unroll
      for (int nt = 0; nt < 2; ++nt)
#pragma unroll
        for (int v = 0; v < 8; ++v)
          ob[(size_t)(rhalf + v) * CC + nt * 16 + nl] = acc[m][nt][v];
    }
  }
}

// ---------------------------------------------------------------------------
extern "C" void kernel_launch(void* const* d_in, const int* in_sizes, int n_in,
                              void* d_out, int out_size, void* d_ws, size_t ws_size,
                              hipStream_t stream) {
  (void)in_sizes; (void)n_in; (void)out_size; (void)ws_size;
  const float* emb1 = (const float*)d_in[0];
  const float* emb2 = (const float*)d_in[1];
  const float* Wq1 = (const float*)d_in[2];
  const float* Wq2 = (const float*)d_in[3];
  const float* Wout1 = (const float*)d_in[4];
  const float* Wout2 = (const float*)d_in[5];
  float* out = (float*)d_out;
  float* ws = (float*)d_ws;

  pool_kernel<<<1024, 256, 0, stream>>>(emb1, emb2, ws);
  proj_kernel<<<32, 32, 0, stream>>>(Wq1, Wq2, ws);
  attn_kernel<<<16, 64, 0, stream>>>(Wout1, Wout2, ws,
                                     out + (size_t)3 * BB * NN * CC);
  gemm_kernel<<<1024, 256, 0, stream>>>(emb1, emb2, ws, out);
}